// EquivariantInterface_42631845380937
// MI455X (gfx1250) — compile-verified
//
#include <hip/hip_runtime.h>
#include <hip/hip_bf16.h>
#include <math.h>

typedef __attribute__((ext_vector_type(2))) float f32x2;
typedef __attribute__((ext_vector_type(8))) float f32x8;

#define N_PTS   2048
#define B_TOT   16      // b(4) * k(4)
#define DIM     32
#define KNN     4
#define EPS_F   1e-8f
#define SLOPE_F 0.2f
#define GSEPS_F 1e-6f

// ---------------------------------------------------------------------------
// Kernel 1: x = node + z ; build comp-major X[B][6][N] (for WMMA tiles),
// point-major pts[B][N][6] (for neighbor gathers), xx[B][N] = ||x||^2.
// comp = c*3 + axis  (matches reference reshape (C,3)->6 then transpose).
// ---------------------------------------------------------------------------
__global__ void eqi_prep_kernel(const float* __restrict__ node,
                                const float* __restrict__ z,
                                float* __restrict__ X,
                                float* __restrict__ pts,
                                float* __restrict__ xx) {
  int t = blockIdx.x * blockDim.x + threadIdx.x;
  if (t >= B_TOT * N_PTS) return;
  int B = t >> 11, n = t & (N_PTS - 1);
  int b = B >> 2, kk = B & 3;
  const float* nf = node + (size_t)(b * N_PTS + n) * 6;              // [axis][c]
  const float* zf = z + (size_t)((b * 4 + kk) * N_PTS + n) * 6;
  float s = 0.f;
#pragma unroll
  for (int c = 0; c < 2; ++c)
#pragma unroll
    for (int ax = 0; ax < 3; ++ax) {
      float v = nf[ax * 2 + c] + zf[ax * 2 + c];
      int comp = c * 3 + ax;
      X[((size_t)B * 6 + comp) * N_PTS + n] = v;
      pts[((size_t)B * N_PTS + n) * 6 + comp] = v;
      s += v * v;
    }
  xx[(size_t)B * N_PTS + n] = s;
}

// ---------------------------------------------------------------------------
// Kernel 2: WMMA f32 16x16x4 Gram matrix + streaming per-row top-4.
// One wave per (cloud, 16-row tile): 16*128 = 2048 workgroups of 32 threads.
// neg_dist[i][j] = 2*inner - xx[i] - xx[j]; keep 4 largest per row.
// ---------------------------------------------------------------------------
__global__ void eqi_dist_topk_kernel(const float* __restrict__ X,
                                     const float* __restrict__ xx,
                                     int* __restrict__ knn_idx) {
  __shared__ float nd[16][17];                    // 16x16 tile, padded
  int wg   = blockIdx.x;
  int B    = wg >> 7;                             // cloud
  int i0   = (wg & 127) * 16;                     // row tile base
  int lane = threadIdx.x;
  int mi   = lane & 15;
  bool hi  = lane >= 16;
  const float* Xb  = X + (size_t)B * 6 * N_PTS;
  const float* xxb = xx + (size_t)B * N_PTS;

  // A tile (rows i0..i0+15), K split: wmma#0 -> comps 0..3, wmma#1 -> comps 4,5
  float a0 = Xb[0 * N_PTS + i0 + mi], a1 = Xb[1 * N_PTS + i0 + mi];
  float a2 = Xb[2 * N_PTS + i0 + mi], a3 = Xb[3 * N_PTS + i0 + mi];
  float a4 = Xb[4 * N_PTS + i0 + mi], a5 = Xb[5 * N_PTS + i0 + mi];
  f32x2 A0, A1;
  A0.x = hi ? a2 : a0;   A0.y = hi ? a3 : a1;
  A1.x = hi ? 0.f : a4;  A1.y = hi ? 0.f : a5;

  float xxrow[8];
#pragma unroll
  for (int r = 0; r < 8; ++r) xxrow[r] = xxb[i0 + r + (hi ? 8 : 0)];

  float tv0 = -INFINITY, tv1 = -INFINITY, tv2 = -INFINITY, tv3 = -INFINITY;
  int   ti0 = 0, ti1 = 0, ti2 = 0, ti3 = 0;

  for (int jt = 0; jt < N_PTS / 16; ++jt) {
    int j0 = jt * 16;
    int jm = j0 + mi;
    if (jt + 1 < N_PTS / 16)                       // hint next column tile
      __builtin_prefetch(&Xb[j0 + 16 + mi], 0, 1);

    float c0 = Xb[0 * N_PTS + jm], c1 = Xb[1 * N_PTS + jm];
    float c2 = Xb[2 * N_PTS + jm], c3 = Xb[3 * N_PTS + jm];
    float c4 = Xb[4 * N_PTS + jm], c5 = Xb[5 * N_PTS + jm];
    f32x2 Bt0, Bt1;
    Bt0.x = hi ? c2 : c0;   Bt0.y = hi ? c3 : c1;
    Bt1.x = hi ? 0.f : c4;  Bt1.y = hi ? 0.f : c5;

    f32x8 acc = {0.f, 0.f, 0.f, 0.f, 0.f, 0.f, 0.f, 0.f};
    acc = __builtin_amdgcn_wmma_f32_16x16x4_f32(false, A0, false, Bt0,
                                                (short)0, acc, false, false);
    acc = __builtin_amdgcn_wmma_f32_16x16x4_f32(false, A1, false, Bt1,
                                                (short)0, acc, false, false);

    float xxcol = xxb[jm];
#pragma unroll
    for (int r = 0; r < 8; ++r) {
      int row = r + (hi ? 8 : 0);
      nd[row][mi] = 2.f * acc[r] - xxrow[r] - xxcol;
    }
    __syncthreads();

    if (lane < 16) {                               // lane == row within tile
#pragma unroll
      for (int c = 0; c < 16; ++c) {
        float v = nd[lane][c];
        int   j = j0 + c;
        if (v > tv0) { tv3=tv2; ti3=ti2; tv2=tv1; ti2=ti1; tv1=tv0; ti1=ti0; tv0=v; ti0=j; }
        else if (v > tv1) { tv3=tv2; ti3=ti2; tv2=tv1; ti2=ti1; tv1=v; ti1=j; }
        else if (v > tv2) { tv3=tv2; ti3=ti2; tv2=v; ti2=j; }
        else if (v > tv3) { tv3=v; ti3=j; }
      }
    }
    __syncthreads();
  }

  if (lane < 16) {
    size_t o = ((size_t)B * N_PTS + i0 + lane) * KNN;
    knn_idx[o + 0] = ti0; knn_idx[o + 1] = ti1;
    knn_idx[o + 2] = ti2; knn_idx[o + 3] = ti3;
  }
}

// ---------------------------------------------------------------------------
// Kernel 3: VN linear-leakyReLU (per-thread, 32x4 weights) then the two
// 32x32 channel GEMMs (Wp1 @ H, Wp3 @ X1) on WMMA f32 16x16x4.
// Block = 256 threads = 8 points x 32 channels = 8 waves.
// LDS layout: col = (pt*KNN + j)*3 + ax  for H/D1 ; col = pt*3 + ax for X1/D3.
// ---------------------------------------------------------------------------
#define NC1 (8 * KNN * 3)   // 96 columns for stage-2 GEMM
#define NC2 32              // 24 used + 8 pad columns for stage-3 GEMM

__global__ void eqi_feat_kernel(const float* __restrict__ pts,
                                const int* __restrict__ knn_idx,
                                const float* __restrict__ W1f,
                                const float* __restrict__ W1d,
                                const float* __restrict__ Wp1,
                                const float* __restrict__ Wp3,
                                float* __restrict__ x1,
                                float* __restrict__ dotn) {
  __shared__ float hL[DIM][NC1];    // H  : [in-ch][col]
  __shared__ float d1L[DIM][NC1];   // D1 = Wp1 @ H
  __shared__ float x1L[DIM][NC2];   // X1 : [in-ch][col]
  __shared__ float d3L[DIM][NC2];   // D3 = Wp3 @ X1
  int pt = threadIdx.x >> 5;        // 0..7  (== wave id)
  int ch = threadIdx.x & 31;
  size_t gp = (size_t)blockIdx.x * 8 + pt;        // global point over B*N
  int B = (int)(gp >> 11), n = (int)(gp & (N_PTS - 1));

  // ---- stage 1: h per neighbor (per-thread; 32x4 weight rows) -------------
  float Wf[4], Wd[4], ctr6[6];
#pragma unroll
  for (int c = 0; c < 4; ++c) { Wf[c] = W1f[ch * 4 + c]; Wd[c] = W1d[ch * 4 + c]; }
#pragma unroll
  for (int i = 0; i < 6; ++i) ctr6[i] = pts[gp * 6 + i];

#pragma unroll
  for (int j = 0; j < KNN; ++j) {
    int nb = knn_idx[gp * KNN + j];
    const float* nbp = pts + ((size_t)B * N_PTS + nb) * 6;
    float p[3] = {0.f, 0.f, 0.f}, dd[3] = {0.f, 0.f, 0.f};
#pragma unroll
    for (int c = 0; c < 2; ++c)
#pragma unroll
      for (int ax = 0; ax < 3; ++ax) {
        float e = nbp[c * 3 + ax] - ctr6[c * 3 + ax];
        p[ax]  += Wf[c] * e;  dd[ax] += Wd[c] * e;
      }
#pragma unroll
    for (int c = 0; c < 2; ++c)
#pragma unroll
      for (int ax = 0; ax < 3; ++ax) {
        float e = ctr6[c * 3 + ax];
        p[ax]  += Wf[2 + c] * e;  dd[ax] += Wd[2 + c] * e;
      }
    float dot = p[0]*dd[0] + p[1]*dd[1] + p[2]*dd[2];
    float d2  = dd[0]*dd[0] + dd[1]*dd[1] + dd[2]*dd[2];
    float inv = dot / (d2 + EPS_F);
    int base = (pt * KNN + j) * 3;
#pragma unroll
    for (int ax = 0; ax < 3; ++ax) {
      float neg = p[ax] - inv * dd[ax];
      float sel = (dot >= 0.f) ? p[ax] : neg;
      hL[ch][base + ax] = SLOPE_F * p[ax] + (1.f - SLOPE_F) * sel;
    }
  }
  __syncthreads();

  // ---- stage 2a: D1 = Wp1(32x32) @ H(32x96) on WMMA -----------------------
  // 12 wave-tasks: mt in {0,1} (out-ch tile), nt in {0..5} (col tile).
  {
    int lane = threadIdx.x & 31;
    int mi = lane & 15;
    bool hi = lane >= 16;
    for (int task = pt; task < 12; task += 8) {
      int mt = task & 1, nt = task >> 1;
      int col = nt * 16 + mi;
      int m   = mt * 16 + mi;
      f32x8 acc = {0.f, 0.f, 0.f, 0.f, 0.f, 0.f, 0.f, 0.f};
#pragma unroll
      for (int ks = 0; ks < 8; ++ks) {
        f32x2 A, Bv;
        A.x  = Wp1[m * DIM + ks * 4 + (hi ? 2 : 0)];
        A.y  = Wp1[m * DIM + ks * 4 + (hi ? 3 : 1)];
        Bv.x = hL[ks * 4 + (hi ? 2 : 0)][col];
        Bv.y = hL[ks * 4 + (hi ? 3 : 1)][col];
        acc = __builtin_amdgcn_wmma_f32_16x16x4_f32(false, A, false, Bv,
                                                    (short)0, acc, false, false);
      }
#pragma unroll
      for (int r = 0; r < 8; ++r)
        d1L[mt * 16 + r + (hi ? 8 : 0)][col] = acc[r];
    }
  }
  __syncthreads();

  // ---- stage 2b: per (pt,ch): dot over axis, argmax over j, gather --------
  float best = -INFINITY; int bj = 0;
#pragma unroll
  for (int j = 0; j < KNN; ++j) {
    int base = (pt * KNN + j) * 3;
    float dt = hL[ch][base + 0] * d1L[ch][base + 0]
             + hL[ch][base + 1] * d1L[ch][base + 1]
             + hL[ch][base + 2] * d1L[ch][base + 2];
    if (dt > best) { best = dt; bj = j; }
  }
  float xv[3];
  int bbase = (pt * KNN + bj) * 3;
#pragma unroll
  for (int ax = 0; ax < 3; ++ax) {
    xv[ax] = hL[ch][bbase + ax];
    x1L[ch][pt * 3 + ax] = xv[ax];
  }
  if (pt == 0) {                       // zero the pad columns 24..31
#pragma unroll
    for (int c = 24; c < NC2; ++c) x1L[ch][c] = 0.f;
  }
  __syncthreads();

  // ---- stage 3a: D3 = Wp3(32x32) @ X1(32x32) on WMMA ----------------------
  // 4 wave-tasks: mt in {0,1}, nt in {0,1}.
  if (pt < 4) {
    int lane = threadIdx.x & 31;
    int mi = lane & 15;
    bool hi = lane >= 16;
    int mt = pt & 1, nt = pt >> 1;
    int col = nt * 16 + mi;
    int m   = mt * 16 + mi;
    f32x8 acc = {0.f, 0.f, 0.f, 0.f, 0.f, 0.f, 0.f, 0.f};
#pragma unroll
    for (int ks = 0; ks < 8; ++ks) {
      f32x2 A, Bv;
      A.x  = Wp3[m * DIM + ks * 4 + (hi ? 2 : 0)];
      A.y  = Wp3[m * DIM + ks * 4 + (hi ? 3 : 1)];
      Bv.x = x1L[ks * 4 + (hi ? 2 : 0)][col];
      Bv.y = x1L[ks * 4 + (hi ? 3 : 1)][col];
      acc = __builtin_amdgcn_wmma_f32_16x16x4_f32(false, A, false, Bv,
                                                  (short)0, acc, false, false);
    }
#pragma unroll
    for (int r = 0; r < 8; ++r)
      d3L[mt * 16 + r + (hi ? 8 : 0)][col] = acc[r];
  }
  __syncthreads();

  // ---- stage 3b: dotn + global writes -------------------------------------
  float dt = xv[0] * d3L[ch][pt * 3 + 0]
           + xv[1] * d3L[ch][pt * 3 + 1]
           + xv[2] * d3L[ch][pt * 3 + 2];
#pragma unroll
  for (int ax = 0; ax < 3; ++ax)
    x1[(((size_t)B * DIM + ch) * 3 + ax) * N_PTS + n] = xv[ax];
  dotn[((size_t)B * DIM + ch) * N_PTS + n] = dt;
}

// ---------------------------------------------------------------------------
// Kernel 4: global argmax over N per (B,ch), 3x32 head, Gram-Schmidt, det.
// One block (32 waves) per cloud; wave w owns channel w.
// ---------------------------------------------------------------------------
__global__ void eqi_pool_final_kernel(const float* __restrict__ x1,
                                      const float* __restrict__ dotn,
                                      const float* __restrict__ Wh2,
                                      const float* __restrict__ noise,
                                      float* __restrict__ out) {
  __shared__ float pooled[DIM][3];
  int B = blockIdx.x;
  int ch = threadIdx.x >> 5;
  int lane = threadIdx.x & 31;
  const float* dn = dotn + ((size_t)B * DIM + ch) * N_PTS;

  float best = -INFINITY; int bi = 0x7fffffff;
  for (int n = lane; n < N_PTS; n += 32) {
    float v = dn[n];
    if (v > best) { best = v; bi = n; }
  }
#pragma unroll
  for (int off = 16; off > 0; off >>= 1) {
    float ov = __shfl_xor(best, off, 32);
    int   oi = __shfl_xor(bi, off, 32);
    if (ov > best || (ov == best && oi < bi)) { best = ov; bi = oi; }
  }
  if (lane == 0) {
#pragma unroll
    for (int ax = 0; ax < 3; ++ax)
      pooled[ch][ax] = x1[(((size_t)B * DIM + ch) * 3 + ax) * N_PTS + bi];
  }
  __syncthreads();

  if (threadIdx.x == 0) {
    // M[axis][o] = (Wh2 @ pooled)^T + GS_EPS * noise
    float M[3][3];
#pragma unroll
    for (int o = 0; o < 3; ++o)
#pragma unroll
      for (int ax = 0; ax < 3; ++ax) {
        float s = 0.f;
        for (int c = 0; c < DIM; ++c) s += Wh2[o * DIM + c] * pooled[c][ax];
        M[ax][o] = s + GSEPS_F * noise[B * 9 + ax * 3 + o];
      }
    float v1[3] = {M[0][0], M[1][0], M[2][0]};
    float v2[3] = {M[0][1], M[1][1], M[2][1]};
    float v3[3] = {M[0][2], M[1][2], M[2][2]};
    float e1[3], e2[3], e3[3], u[3];
    float n1 = sqrtf(v1[0]*v1[0] + v1[1]*v1[1] + v1[2]*v1[2]) + EPS_F;
    for (int i = 0; i < 3; ++i) e1[i] = v1[i] / n1;
    float d12 = e1[0]*v2[0] + e1[1]*v2[1] + e1[2]*v2[2];
    for (int i = 0; i < 3; ++i) u[i] = v2[i] - d12 * e1[i];
    float n2 = sqrtf(u[0]*u[0] + u[1]*u[1] + u[2]*u[2]) + EPS_F;
    for (int i = 0; i < 3; ++i) e2[i] = u[i] / n2;
    float d13 = e1[0]*v3[0] + e1[1]*v3[1] + e1[2]*v3[2];
    float d23 = e2[0]*v3[0] + e2[1]*v3[1] + e2[2]*v3[2];
    for (int i = 0; i < 3; ++i) u[i] = v3[i] - d13 * e1[i] - d23 * e2[i];
    float n3 = sqrtf(u[0]*u[0] + u[1]*u[1] + u[2]*u[2]) + EPS_F;
    for (int i = 0; i < 3; ++i) e3[i] = u[i] / n3;
    // det(R) with columns e1,e2,e3 = e1 . (e2 x e3)
    float cx = e2[1]*e3[2] - e2[2]*e3[1];
    float cy = e2[2]*e3[0] - e2[0]*e3[2];
    float cz = e2[0]*e3[1] - e2[1]*e3[0];
    float det = e1[0]*cx + e1[1]*cy + e1[2]*cz;
    for (int r = 0; r < 3; ++r) {
      out[B * 9 + r * 3 + 0] = e1[r] * det;
      out[B * 9 + r * 3 + 1] = e2[r];
      out[B * 9 + r * 3 + 2] = e3[r];
    }
  }
}

// ---------------------------------------------------------------------------
extern "C" void kernel_launch(void* const* d_in, const int* in_sizes, int n_in,
                              void* d_out, int out_size, void* d_ws, size_t ws_size,
                              hipStream_t stream) {
  const float* node  = (const float*)d_in[0];
  // d_in[1] = edge_features : unused by the reference
  const float* W1f   = (const float*)d_in[4];
  const float* W1d   = (const float*)d_in[5];
  const float* Wp1   = (const float*)d_in[6];
  const float* Wp3   = (const float*)d_in[7];
  const float* Wh2   = (const float*)d_in[8];
  const float* z     = (const float*)d_in[9];
  const float* noise = (const float*)d_in[10];
  float* out = (float*)d_out;

  // workspace carve-out (floats)
  float* X    = (float*)d_ws;                                  // 16*6*2048
  float* pts  = X + (size_t)B_TOT * 6 * N_PTS;                 // 16*2048*6
  float* xx   = pts + (size_t)B_TOT * N_PTS * 6;               // 16*2048
  float* x1   = xx + (size_t)B_TOT * N_PTS;                    // 16*32*3*2048
  float* dotn = x1 + (size_t)B_TOT * DIM * 3 * N_PTS;          // 16*32*2048
  int* knn    = (int*)(dotn + (size_t)B_TOT * DIM * N_PTS);    // 16*2048*4

  eqi_prep_kernel<<<(B_TOT * N_PTS + 255) / 256, 256, 0, stream>>>(
      node, z, X, pts, xx);
  eqi_dist_topk_kernel<<<B_TOT * (N_PTS / 16), 32, 0, stream>>>(X, xx, knn);
  eqi_feat_kernel<<<(B_TOT * N_PTS) / 8, 256, 0, stream>>>(
      pts, knn, W1f, W1d, Wp1, Wp3, x1, dotn);
  eqi_pool_final_kernel<<<B_TOT, 1024, 0, stream>>>(x1, dotn, Wh2, noise, out);
}